// HMM_6854767804555
// MI455X (gfx1250) — compile-verified
//
#include <hip/hip_runtime.h>

// HMM forward-backward (scaled domain) for B=256, T=8192, K=8, binary obs.
// d_out = gamma (B*T*K f32) || loglik (B f32).
// alpha_t = em_t (*) (T^T alpha_{t-1}) batched 16 seqs/wave via
// V_WMMA_F32_16X16X4_F32 (K=8 as two K=4 chunks). Latency-optimized:
// v_permlanex16 cross-half swaps (no LDS), v_rcp+Newton normalization folded
// into the B operand, unnormalized alpha staged in the gamma slot of d_out.

#define HMM_B 256
#define HMM_T 8192
#define HMM_K 8
#define HMM_EPS 1e-8f
#define HMM_LN2 0.6931471805599453f

typedef __attribute__((ext_vector_type(2))) float v2f;
typedef __attribute__((ext_vector_type(8))) float v8f;

// v_rcp_f32 + one Newton step: ~1 ulp rel error, 3 dependent ops vs ~10 for IEEE div.
__device__ __forceinline__ float fast_rcp(float x) {
    float r = __builtin_amdgcn_rcpf(x);
    float e = __builtin_fmaf(-x, r, 1.0f);
    return __builtin_fmaf(r, e, r);
}

// lane i <-> lane i^16 swap via v_permlanex16_b32 (VALU, no LDS round trip).
__device__ __forceinline__ float crosshalf(float x) {
    unsigned u = __builtin_bit_cast(unsigned, x);
    u = __builtin_amdgcn_permlanex16(u, u, 0x76543210u, 0xFEDCBA98u, false, false);
    return __builtin_bit_cast(float, u);
}

// ---------------------------------------------------------------------------
// Kernel 1: pack binary obs (B,T) int32 -> bitmask words (B, T/32) uint32.
// ---------------------------------------------------------------------------
__global__ __launch_bounds__(256) void hmm_pack_kernel(
    const int* __restrict__ obs, unsigned* __restrict__ bits)
{
    const int wave = (blockIdx.x * blockDim.x + threadIdx.x) >> 5;
    const int lane = threadIdx.x & 31;
    const long base = (long)wave * 1024;
    unsigned keep = 0;
#pragma unroll 4
    for (int i = 0; i < 32; ++i) {
        int v = obs[base + i * 32 + lane];
        unsigned m = (unsigned)__ballot(v != 0);
        if (i == lane) keep = m;   // lane L keeps word L -> coalesced store
    }
    bits[(size_t)wave * 32 + lane] = keep;
}

// ---------------------------------------------------------------------------
// Forward: one wave per 16 sequences. State a = unnormalized alpha in WMMA
// C/D layout (vgpr m = state m, lanes 0-15 = sequences). The per-column
// normalizer inv is folded into the B operand each step.
// ---------------------------------------------------------------------------
__global__ __launch_bounds__(32) void hmm_forward_kernel(
    const float* __restrict__ start, const float* __restrict__ trans,
    const float* __restrict__ emission, const unsigned* __restrict__ bits,
    float* __restrict__ alpha_out, float* __restrict__ loglik)
{
    const int  lane = threadIdx.x;
    const int  n    = lane & 15;
    const bool lo   = lane < 16;
    const int  b    = blockIdx.x * 16 + n;

    // ---- normalized parameters ----
    float em1[8], em0[8];
#pragma unroll
    for (int k = 0; k < 8; ++k) {
        float e = emission[k];
        e = fminf(fmaxf(e, HMM_EPS), 1.0f - HMM_EPS);
        em1[k] = e; em0[k] = 1.0f - e;
    }
    float st[8]; float ssum = 0.0f;
#pragma unroll
    for (int k = 0; k < 8; ++k) { st[k] = start[k] + HMM_EPS; ssum += st[k]; }
#pragma unroll
    for (int k = 0; k < 8; ++k) st[k] /= ssum;

    float rs[8];
#pragma unroll
    for (int j = 0; j < 8; ++j) {
        float s = 0.0f;
#pragma unroll
        for (int k = 0; k < 8; ++k) s += trans[j * 8 + k] + HMM_EPS;
        rs[j] = s;
    }

    // ---- constant A = Trans^T (M = out state, K = in state) ----
    // A 16x4 layout: lanes 0-15 -> K=0 (v0), K=1 (v1); lanes 16-31 -> K=2,3.
    const int kb = lo ? 0 : 2;
    const int i  = n;
    v2f A0, A1;
    A0.x = (i < 8) ? (trans[(kb + 0) * 8 + i] + HMM_EPS) / rs[kb + 0] : 0.0f;
    A0.y = (i < 8) ? (trans[(kb + 1) * 8 + i] + HMM_EPS) / rs[kb + 1] : 0.0f;
    A1.x = (i < 8) ? (trans[(kb + 4) * 8 + i] + HMM_EPS) / rs[kb + 4] : 0.0f;
    A1.y = (i < 8) ? (trans[(kb + 5) * 8 + i] + HMM_EPS) / rs[kb + 5] : 0.0f;

    // ---- t = 0 ----
    unsigned w  = bits[(size_t)b * (HMM_T / 32)];
    int      ob = (int)(w & 1u);
    v8f a;                                   // unnormalized alpha
#pragma unroll
    for (int m = 0; m < 8; ++m) a[m] = st[m] * (ob ? em1[m] : em0[m]);
    float s = a[0] + a[1] + a[2] + a[3] + a[4] + a[5] + a[6] + a[7];
    {  float sx = crosshalf(s); s = lo ? s : sx; }
    float inv = fast_rcp(s);
    float lc  = __builtin_amdgcn_logf(s);    // log2; scaled by ln2 at the end
    if (lo) *(v8f*)(alpha_out + ((size_t)b * HMM_T) * 8) = a;

    const v8f vzero = {};
    for (int t = 1; t < HMM_T; ++t) {
        if ((t & 31) == 0) w = bits[(size_t)b * (HMM_T / 32) + (t >> 5)];
        ob = (int)((w >> (t & 31)) & 1u);

        // B_t = inv * alpha_un (D layout -> B 4x16 layout via cross-half swap)
        float x2 = crosshalf(a[2]);
        float x3 = crosshalf(a[3]);
        float x6 = crosshalf(a[6]);
        float x7 = crosshalf(a[7]);
        v2f B0, B1;
        B0.x = (lo ? a[0] : x2) * inv;  B0.y = (lo ? a[1] : x3) * inv;
        B1.x = (lo ? a[4] : x6) * inv;  B1.y = (lo ? a[5] : x7) * inv;

        v8f d = __builtin_amdgcn_wmma_f32_16x16x4_f32(
            false, A0, false, B0, (short)0, vzero, false, false);
        d = __builtin_amdgcn_wmma_f32_16x16x4_f32(
            false, A1, false, B1, (short)0, d, false, false);

#pragma unroll
        for (int m = 0; m < 8; ++m) a[m] = d[m] * (ob ? em1[m] : em0[m]);
        s = a[0] + a[1] + a[2] + a[3] + a[4] + a[5] + a[6] + a[7];
        float sx = crosshalf(s);            // upper half rows of D are zero
        s = lo ? s : sx;                    // both halves: this column's sum
        inv = fast_rcp(s);
        lc += __builtin_amdgcn_logf(s);
        if (lo) *(v8f*)(alpha_out + ((size_t)b * HMM_T + t) * 8) = a;
    }
    if (lo) loglik[b] = lc * HMM_LN2;
}

// ---------------------------------------------------------------------------
// Backward: d = (scaled) beta in D layout; per step B = inv * (em (*) d),
// d_new = Trans * B; gamma_t = normalize(alpha_staged (*) d_new) in place.
// ---------------------------------------------------------------------------
__global__ __launch_bounds__(32) void hmm_backward_kernel(
    const float* __restrict__ trans, const float* __restrict__ emission,
    const unsigned* __restrict__ bits, float* __restrict__ gamma)
{
    const int  lane = threadIdx.x;
    const int  n    = lane & 15;
    const bool lo   = lane < 16;
    const int  b    = blockIdx.x * 16 + n;

    float em1[8], em0[8];
#pragma unroll
    for (int k = 0; k < 8; ++k) {
        float e = emission[k];
        e = fminf(fmaxf(e, HMM_EPS), 1.0f - HMM_EPS);
        em1[k] = e; em0[k] = 1.0f - e;
    }
    float rs[8];
#pragma unroll
    for (int j = 0; j < 8; ++j) {
        float s = 0.0f;
#pragma unroll
        for (int k = 0; k < 8; ++k) s += trans[j * 8 + k] + HMM_EPS;
        rs[j] = s;
    }
    // ---- constant A = Trans (M = j, K = k contraction) ----
    const int kb = lo ? 0 : 2;
    const int i  = n;
    v2f A0, A1;
    A0.x = (i < 8) ? (trans[i * 8 + (kb + 0)] + HMM_EPS) / rs[i] : 0.0f;
    A0.y = (i < 8) ? (trans[i * 8 + (kb + 1)] + HMM_EPS) / rs[i] : 0.0f;
    A1.x = (i < 8) ? (trans[i * 8 + (kb + 4)] + HMM_EPS) / rs[i] : 0.0f;
    A1.y = (i < 8) ? (trans[i * 8 + (kb + 5)] + HMM_EPS) / rs[i] : 0.0f;

    // beta(T-1) uniform = 1; gamma(T-1) = normalize(alpha_staged)
    v8f d;
#pragma unroll
    for (int m = 0; m < 8; ++m) d[m] = 1.0f;
    if (lo) {
        float* p = gamma + ((size_t)b * HMM_T + (HMM_T - 1)) * 8;
        v8f al = *(v8f*)p;
        float gs = al[0]+al[1]+al[2]+al[3]+al[4]+al[5]+al[6]+al[7];
        float gi = fast_rcp(gs);
        v8f g;
#pragma unroll
        for (int m = 0; m < 8; ++m) g[m] = al[m] * gi;
        *(v8f*)p = g;
    }

    const v8f vzero = {};
    unsigned w = 0;
    for (int t = HMM_T - 2; t >= 0; --t) {
        const int tt = t + 1;
        if ((tt & 31) == 31) w = bits[(size_t)b * (HMM_T / 32) + (tt >> 5)];
        int ob = (int)((w >> (tt & 31)) & 1u);

        // normalizer from previous beta (any positive per-column scale is OK)
        float s = d[0]+d[1]+d[2]+d[3]+d[4]+d[5]+d[6]+d[7];
        float sx = crosshalf(s);
        s = lo ? s : sx;
        float inv = fast_rcp(s);

        v8f u;
#pragma unroll
        for (int m = 0; m < 8; ++m) u[m] = d[m] * (ob ? em1[m] : em0[m]);
        float x2 = crosshalf(u[2]);
        float x3 = crosshalf(u[3]);
        float x6 = crosshalf(u[6]);
        float x7 = crosshalf(u[7]);
        v2f B0, B1;
        B0.x = (lo ? u[0] : x2) * inv;  B0.y = (lo ? u[1] : x3) * inv;
        B1.x = (lo ? u[4] : x6) * inv;  B1.y = (lo ? u[5] : x7) * inv;

        d = __builtin_amdgcn_wmma_f32_16x16x4_f32(
            false, A0, false, B0, (short)0, vzero, false, false);
        d = __builtin_amdgcn_wmma_f32_16x16x4_f32(
            false, A1, false, B1, (short)0, d, false, false);

        if (lo) {
            float* p = gamma + ((size_t)b * HMM_T + t) * 8;
            v8f al = *(v8f*)p;
            v8f g;
            float gs = 0.0f;
#pragma unroll
            for (int m = 0; m < 8; ++m) { g[m] = al[m] * d[m]; gs += g[m]; }
            float gi = fast_rcp(gs);
#pragma unroll
            for (int m = 0; m < 8; ++m) g[m] *= gi;
            *(v8f*)p = g;
        }
    }
}

// ---------------------------------------------------------------------------
extern "C" void kernel_launch(void* const* d_in, const int* in_sizes, int n_in,
                              void* d_out, int out_size, void* d_ws, size_t ws_size,
                              hipStream_t stream) {
    (void)in_sizes; (void)n_in; (void)out_size; (void)ws_size;
    const float* start    = (const float*)d_in[0];
    const float* trans    = (const float*)d_in[1];
    const float* emission = (const float*)d_in[2];
    const int*   obs      = (const int*)d_in[3];

    float* gamma  = (float*)d_out;
    float* loglik = gamma + (size_t)HMM_B * HMM_T * HMM_K;
    unsigned* bits = (unsigned*)d_ws;     // B*T/32 words = 256 KiB (fits L2)

    hmm_pack_kernel<<<256, 256, 0, stream>>>(obs, bits);
    hmm_forward_kernel<<<HMM_B / 16, 32, 0, stream>>>(
        start, trans, emission, bits, gamma, loglik);
    hmm_backward_kernel<<<HMM_B / 16, 32, 0, stream>>>(
        trans, emission, bits, gamma);
}